// KalmanFilter_28827820490983
// MI455X (gfx1250) — compile-verified
//
#include <hip/hip_runtime.h>

typedef __attribute__((ext_vector_type(2))) float v2f;
typedef __attribute__((ext_vector_type(8))) float v8f;

#define G_N 1024
#define T_N 256
#define M_N 16
#define S_N 32

// ---- WMMA f32 16x16x4 fragment helpers (wave32) ----
// A 16x4: lanes 0-15 -> M=lane, K={0,1}; lanes 16-31 -> M=lane-16, K={2,3}
__device__ __forceinline__ v2f frag_a(const float* S, int ld, int r0, int k0, int lane) {
  int m = r0 + (lane & 15);
  int k = k0 + ((lane >> 4) << 1);
  v2f a;
  a.x = S[m * ld + k];
  a.y = S[m * ld + k + 1];
  return a;
}
// B 4x16: row striped across lanes within a VGPR; VGPR j holds K=j+2*half
__device__ __forceinline__ v2f frag_b(const float* S, int ld, int k0, int c0, int lane) {
  int n = c0 + (lane & 15);
  int k = k0 + ((lane >> 4) << 1);
  v2f b;
  b.x = S[k * ld + n];
  b.y = S[(k + 1) * ld + n];
  return b;
}
// B = X^T with X row-major [n][k]
__device__ __forceinline__ v2f frag_bt(const float* X, int ld, int k0, int c0, int lane) {
  int n = c0 + (lane & 15);
  int k = k0 + ((lane >> 4) << 1);
  v2f b;
  b.x = X[n * ld + k];
  b.y = X[n * ld + k + 1];
  return b;
}
// C/D 16x16 f32: VGPR v -> rows v (lanes 0-15) and v+8 (lanes 16-31)
__device__ __forceinline__ v8f frag_c_load(const float* S, int ld, int r0, int c0, int lane) {
  int half = lane >> 4;
  int c = c0 + (lane & 15);
  v8f acc;
#pragma unroll
  for (int v = 0; v < 8; ++v) acc[v] = S[(r0 + v + 8 * half) * ld + c];
  return acc;
}
__device__ __forceinline__ void frag_c_store(float* S, int ld, int r0, int c0, int lane, v8f acc) {
  int half = lane >> 4;
  int c = c0 + (lane & 15);
#pragma unroll
  for (int v = 0; v < 8; ++v) S[(r0 + v + 8 * half) * ld + c] = acc[v];
}

__device__ __forceinline__ v8f wmma4(v2f a, v2f b, v8f c) {
  // D = A(16x4,f32) * B(4x16,f32) + C(16x16,f32); full fp32 precision
  return __builtin_amdgcn_wmma_f32_16x16x4_f32(false, a, false, b, (short)0, c, false, false);
}

__global__ __launch_bounds__(32) void kalman1250(
    const float* __restrict__ inp, const float* __restrict__ Fg,
    const float* __restrict__ Qg, const float* __restrict__ Hg,
    const float* __restrict__ Rg, const float* __restrict__ m0g,
    const float* __restrict__ P0g, float* __restrict__ out_mu,
    float* __restrict__ out_sig) {
  __shared__ float sH[M_N * S_N];     // H   16x32
  __shared__ float sHt[S_N * M_N];    // H^T 32x16
  __shared__ float sF[S_N * S_N];     // F   32x32
  __shared__ float sFt[S_N * S_N];    // F^T 32x32
  __shared__ float sQ[S_N * S_N];
  __shared__ float sR[M_N * M_N];
  __shared__ float sP[S_N * S_N];     // covariance (updated in place)
  __shared__ float sPHt[S_N * M_N];   // P H^T 32x16
  __shared__ float sW[M_N * 2 * M_N]; // [Sig | -I] -> [I | -Sig^{-1}]
  __shared__ float sK[S_N * M_N];     // Kn = -K = PHt * (-SigInv), 32x16
  __shared__ float sT1[S_N * S_N];    // F * P_u
  __shared__ float sm[S_N];           // state mean
  __shared__ float smu[S_N];          // updated mean m_u
  __shared__ float sres[M_N];         // y - H m

  const int lane = (int)threadIdx.x;
  const int g = (int)blockIdx.x;

  // ---- load constants (per block) + per-group init ----
  for (int i = lane; i < M_N * S_N; i += 32) {
    float h = Hg[i];
    sH[i] = h;
    sHt[(i & (S_N - 1)) * M_N + (i >> 5)] = h;   // i = r*32+c -> Ht[c][r]
  }
  for (int i = lane; i < S_N * S_N; i += 32) {
    float f = Fg[i];
    sF[i] = f;
    sFt[(i & (S_N - 1)) * S_N + (i >> 5)] = f;
    sQ[i] = Qg[i];
    sP[i] = P0g[(size_t)g * S_N * S_N + i];
  }
  for (int i = lane; i < M_N * M_N; i += 32) sR[i] = Rg[i];
  sm[lane] = m0g[(size_t)g * S_N + lane];
  __syncthreads();

  for (int t = 0; t < T_N; ++t) {
    const size_t gt = (size_t)g * T_N + t;

    // ---- mu = H m ; resid = y - mu ; emit mu ----
    if (lane < M_N) {
      float acc = 0.f;
#pragma unroll
      for (int s = 0; s < S_N; ++s) acc += sH[lane * S_N + s] * sm[s];
      out_mu[gt * M_N + lane] = acc;
      sres[lane] = inp[gt * M_N + lane] - acc;
    }
    __syncthreads();

    // ---- PHt = P * Ht (32x16), K-dim 32 ----
#pragma unroll
    for (int tr = 0; tr < 2; ++tr) {
      v8f acc = {};
#pragma unroll
      for (int k0 = 0; k0 < S_N; k0 += 4)
        acc = wmma4(frag_a(sP, S_N, tr * 16, k0, lane), frag_b(sHt, M_N, k0, 0, lane), acc);
      frag_c_store(sPHt, M_N, tr * 16, 0, lane, acc);
    }
    __syncthreads();

    // ---- Sig = H * PHt + R (16x16); emit Sig; stage [Sig | -I] ----
    {
      v8f acc = frag_c_load(sR, M_N, 0, 0, lane);
#pragma unroll
      for (int k0 = 0; k0 < S_N; k0 += 4)
        acc = wmma4(frag_a(sH, S_N, 0, k0, lane), frag_b(sPHt, M_N, k0, 0, lane), acc);
      int half = lane >> 4, c = lane & 15;
#pragma unroll
      for (int v = 0; v < 8; ++v) {
        int r = v + 8 * half;
        out_sig[(gt * M_N + r) * M_N + c] = acc[v];
        sW[r * 32 + c] = acc[v];
        sW[r * 32 + 16 + c] = (r == c) ? -1.f : 0.f;
      }
    }
    __syncthreads();

    // ---- Gauss-Jordan on 16x32 augmented system: right half -> -Sig^{-1} ----
    for (int p = 0; p < M_N; ++p) {
      float invp = 1.0f / sW[p * 32 + p];
      float wp = sW[p * 32 + lane] * invp;   // lane owns one column
      __syncthreads();
      sW[p * 32 + lane] = wp;
      __syncthreads();
      for (int r = 0; r < M_N; ++r) {
        if (r == p) continue;
        float fmul = sW[r * 32 + p];         // broadcast load precedes stores
        sW[r * 32 + lane] -= fmul * wp;
      }
      __syncthreads();
    }

    // ---- Kn = PHt * (-SigInv) (32x16), K-dim 16 ----
#pragma unroll
    for (int tr = 0; tr < 2; ++tr) {
      v8f acc = {};
#pragma unroll
      for (int k0 = 0; k0 < M_N; k0 += 4)
        acc = wmma4(frag_a(sPHt, M_N, tr * 16, k0, lane), frag_b(sW, 32, k0, 16, lane), acc);
      frag_c_store(sK, M_N, tr * 16, 0, lane, acc);
    }
    __syncthreads();

    // ---- m_u = m - Kn * resid (Kn = -K so this is m + K resid) ----
    {
      float acc = sm[lane];
#pragma unroll
      for (int j = 0; j < M_N; ++j) acc -= sK[lane * M_N + j] * sres[j];
      smu[lane] = acc;
    }

    // ---- P_u = P + Kn * PHt^T (in place on sP), K-dim 16 ----
#pragma unroll
    for (int tr = 0; tr < 2; ++tr)
#pragma unroll
      for (int tc = 0; tc < 2; ++tc) {
        v8f acc = frag_c_load(sP, S_N, tr * 16, tc * 16, lane);
#pragma unroll
        for (int k0 = 0; k0 < M_N; k0 += 4)
          acc = wmma4(frag_a(sK, M_N, tr * 16, k0, lane),
                      frag_bt(sPHt, M_N, k0, tc * 16, lane), acc);
        frag_c_store(sP, S_N, tr * 16, tc * 16, lane, acc);
      }
    __syncthreads();

    // ---- T1 = F * P_u ; m_n = F * m_u ----
#pragma unroll
    for (int tr = 0; tr < 2; ++tr)
#pragma unroll
      for (int tc = 0; tc < 2; ++tc) {
        v8f acc = {};
#pragma unroll
        for (int k0 = 0; k0 < S_N; k0 += 4)
          acc = wmma4(frag_a(sF, S_N, tr * 16, k0, lane),
                      frag_b(sP, S_N, k0, tc * 16, lane), acc);
        frag_c_store(sT1, S_N, tr * 16, tc * 16, lane, acc);
      }
    {
      float acc = 0.f;
#pragma unroll
      for (int s = 0; s < S_N; ++s) acc += sF[lane * S_N + s] * smu[s];
      sm[lane] = acc;   // same-lane slot; cross-lane reads happen after next barrier
    }
    __syncthreads();

    // ---- P_n = T1 * Ft + Q (into sP), K-dim 32 ----
#pragma unroll
    for (int tr = 0; tr < 2; ++tr)
#pragma unroll
      for (int tc = 0; tc < 2; ++tc) {
        v8f acc = frag_c_load(sQ, S_N, tr * 16, tc * 16, lane);
#pragma unroll
        for (int k0 = 0; k0 < S_N; k0 += 4)
          acc = wmma4(frag_a(sT1, S_N, tr * 16, k0, lane),
                      frag_b(sFt, S_N, k0, tc * 16, lane), acc);
        frag_c_store(sP, S_N, tr * 16, tc * 16, lane, acc);
      }
    __syncthreads();
  }
}

extern "C" void kernel_launch(void* const* d_in, const int* in_sizes, int n_in,
                              void* d_out, int out_size, void* d_ws, size_t ws_size,
                              hipStream_t stream) {
  (void)in_sizes; (void)n_in; (void)out_size; (void)d_ws; (void)ws_size;
  // setup_inputs order: input, F, Q, H, R, m0, P0
  const float* inp = (const float*)d_in[0];
  const float* F   = (const float*)d_in[1];
  const float* Q   = (const float*)d_in[2];
  const float* H   = (const float*)d_in[3];
  const float* R   = (const float*)d_in[4];
  const float* m0  = (const float*)d_in[5];
  const float* P0  = (const float*)d_in[6];
  float* out_mu  = (float*)d_out;                         // (G,T,M)
  float* out_sig = out_mu + (size_t)G_N * T_N * M_N;      // (G,T,M,M)
  kalman1250<<<G_N, 32, 0, stream>>>(inp, F, Q, H, R, m0, P0, out_mu, out_sig);
}